// HistogramMatcher_13408887899066
// MI455X (gfx1250) — compile-verified
//
#include <hip/hip_runtime.h>
#include <math.h>

#define NBINS 256

typedef __attribute__((ext_vector_type(2))) float v2f;
typedef __attribute__((ext_vector_type(8))) float v8f;

__device__ __forceinline__ int iminc(int a, int b) { return a < b ? a : b; }
__device__ __forceinline__ int imaxc(int a, int b) { return a > b ? a : b; }

// ---------------------------------------------------------------------------
// Kernel 0: zero the two 256-bin histograms in workspace (every call).
// ---------------------------------------------------------------------------
__global__ void hm_zero(unsigned* __restrict__ ws) {
    int i = blockIdx.x * blockDim.x + threadIdx.x;
    if (i < 2 * NBINS) ws[i] = 0u;
}

// ---------------------------------------------------------------------------
// Kernel 1: histograms of V = max(r,g,b) after (c+1)*127/255 transform.
// LDS-privatized (wave32-safe: only blockDim/gridDim strides, no lane masks),
// b128 loads: 4 pixels (48B) per thread-iteration.
// ---------------------------------------------------------------------------
__global__ void hm_hist(const float* __restrict__ src, const float* __restrict__ tgt,
                        unsigned* __restrict__ hist, int ngroups) {
    __shared__ unsigned hs[NBINS], ht[NBINS];
    for (int i = threadIdx.x; i < NBINS; i += blockDim.x) { hs[i] = 0u; ht[i] = 0u; }
    __syncthreads();

    int stride = gridDim.x * blockDim.x;
    for (int g = blockIdx.x * blockDim.x + threadIdx.x; g < ngroups; g += stride) {
        size_t base = (size_t)g * 12;
        const float4* ps = (const float4*)(src + base);
        const float4* pt = (const float4*)(tgt + base);

        float4 a = ps[0], b = ps[1], c = ps[2];
        float vmx[4];
        vmx[0] = fmaxf(fmaxf(a.x, a.y), a.z);
        vmx[1] = fmaxf(fmaxf(a.w, b.x), b.y);
        vmx[2] = fmaxf(fmaxf(b.z, b.w), c.x);
        vmx[3] = fmaxf(fmaxf(c.y, c.z), c.w);
#pragma unroll
        for (int k = 0; k < 4; ++k) {
            float v = (vmx[k] + 1.0f) * 127.0f / 255.0f;  // max commutes with increasing affine
            int bin = iminc(imaxc((int)floorf(v * 256.0f), 0), NBINS - 1);
            atomicAdd(&hs[bin], 1u);
        }

        a = pt[0]; b = pt[1]; c = pt[2];
        vmx[0] = fmaxf(fmaxf(a.x, a.y), a.z);
        vmx[1] = fmaxf(fmaxf(a.w, b.x), b.y);
        vmx[2] = fmaxf(fmaxf(b.z, b.w), c.x);
        vmx[3] = fmaxf(fmaxf(c.y, c.z), c.w);
#pragma unroll
        for (int k = 0; k < 4; ++k) {
            float v = (vmx[k] + 1.0f) * 127.0f / 255.0f;
            int bin = iminc(imaxc((int)floorf(v * 256.0f), 0), NBINS - 1);
            atomicAdd(&ht[bin], 1u);
        }
    }
    __syncthreads();
    for (int i = threadIdx.x; i < NBINS; i += blockDim.x) {
        if (hs[i]) atomicAdd(&hist[i], hs[i]);
        if (ht[i]) atomicAdd(&hist[NBINS + i], ht[i]);
    }
}

// ---------------------------------------------------------------------------
// Kernel 2 (single wave32): CDF via WMMA prefix-sum + pxmap construction.
//   cdf(H) = H*U + Lstrict*(H*Ones),  U[k][c]=(k<=c), Lstrict[r][k]=(k<r)
// f32 exact: all partial sums < 2^24.  Fragment layouts per ISA 7.12.2:
//   A 16x4: lane m=L&15, VGPR0/1 = K {0,1} (lanes 0-15) / {2,3} (lanes 16-31)
//   B 4x16: VGPR0/1 = rows {0,1} / {2,3} mirrored across half-waves
//   C/D 16x16: VGPR v = row v (lanes 0-15) / row v+8 (lanes 16-31)
// ---------------------------------------------------------------------------
__device__ __forceinline__ v2f hm_afrag(const float* X, int lane, int kk) {
    int m = lane & 15;
    int kb = 4 * kk + ((lane >> 4) << 1);
    v2f a; a.x = X[m * 16 + kb]; a.y = X[m * 16 + kb + 1];
    return a;
}
__device__ __forceinline__ v2f hm_bfrag(const float* Y, int lane, int kk) {
    int n = lane & 15;
    int kb = 4 * kk + ((lane >> 4) << 1);
    v2f b; b.x = Y[kb * 16 + n]; b.y = Y[(kb + 1) * 16 + n];
    return b;
}
__device__ __forceinline__ void hm_storeC(v8f c, float* X, int lane) {
    int n = lane & 15;
    int mb = (lane >> 4) << 3;
#pragma unroll
    for (int v = 0; v < 8; ++v) X[(mb + v) * 16 + n] = c[v];
}

__global__ __launch_bounds__(32) void hm_cdf_map(const unsigned* __restrict__ hist,
                                                 float* __restrict__ pmout, int npix) {
    __shared__ float Hs[256], Ht[256], Tm[256], cdfS[256], cdfT[256];
    int lane = threadIdx.x;
    for (int i = lane; i < 256; i += 32) {
        Hs[i] = (float)hist[i];
        Ht[i] = (float)hist[256 + i];
    }
    __syncthreads();

    float* Xarr[2] = {Hs, Ht};
    float* Carr[2] = {cdfS, cdfT};
    for (int im = 0; im < 2; ++im) {
        const float* X = Xarr[im];
#if __has_builtin(__builtin_amdgcn_wmma_f32_16x16x4_f32)
        // T = X * Ones   (T[r][*] = row-total of row r)
        v8f t = {};
#pragma unroll
        for (int kk = 0; kk < 4; ++kk) {
            v2f a = hm_afrag(X, lane, kk);
            v2f b; b.x = 1.0f; b.y = 1.0f;
            t = __builtin_amdgcn_wmma_f32_16x16x4_f32(false, a, false, b, (short)0, t, false, false);
        }
        hm_storeC(t, Tm, lane);
        __syncthreads();
        // C = X*U  (row-wise inclusive prefix) + Lstrict*T (preceding-row offsets)
        v8f c = {};
#pragma unroll
        for (int kk = 0; kk < 4; ++kk) {
            v2f a = hm_afrag(X, lane, kk);
            int n = lane & 15;
            int kb = 4 * kk + ((lane >> 4) << 1);
            v2f b;
            b.x = (kb <= n) ? 1.0f : 0.0f;
            b.y = ((kb + 1) <= n) ? 1.0f : 0.0f;
            c = __builtin_amdgcn_wmma_f32_16x16x4_f32(false, a, false, b, (short)0, c, false, false);
        }
#pragma unroll
        for (int kk = 0; kk < 4; ++kk) {
            int m = lane & 15;
            int kb = 4 * kk + ((lane >> 4) << 1);
            v2f a;
            a.x = (kb < m) ? 1.0f : 0.0f;
            a.y = ((kb + 1) < m) ? 1.0f : 0.0f;
            v2f b = hm_bfrag(Tm, lane, kk);
            c = __builtin_amdgcn_wmma_f32_16x16x4_f32(false, a, false, b, (short)0, c, false, false);
        }
        hm_storeC(c, Carr[im], lane);
        __syncthreads();
#else
        // Fallback: lane-0 sequential scan (kept only so the build stays green
        // if the wmma builtin signature differs; should not be taken).
        if (lane == 0) {
            float acc = 0.0f;
            for (int i = 0; i < 256; ++i) { acc += X[i]; Carr[im][i] = acc; }
        }
        __syncthreads();
#endif
    }

    // Normalize: (cdf - cdf[0]) / (npix - 1)   (cdf nondecreasing => min = cdf[0])
    float den = (float)(npix - 1);
    float s0 = cdfS[0], t0 = cdfT[0];
    for (int i = lane; i < 256; i += 32) {
        cdfS[i] = (cdfS[i] - s0) / den;
        cdfT[i] = (cdfT[i] - t0) / den;
    }
    __syncthreads();

    // pxmap = interpolate(dx=cdfT, dy=floating_space, x=cdfS)
    for (int i = lane; i < 256; i += 32) {
        float x = cdfS[i];
        int lo = 0, hi = 256;                 // searchsorted(dx, x, side='right')
        while (lo < hi) {
            int mid = (lo + hi) >> 1;
            if (cdfT[mid] <= x) lo = mid + 1; else hi = mid;
        }
        int ind1 = iminc(lo, 255);
        int ind0 = imaxc(ind1 - 1, 0);
        float dx0 = cdfT[ind0], dx1 = cdfT[ind1];
        float dy0 = (float)ind0 / 255.0f, dy1 = (float)ind1 / 255.0f;
        float d = dx1 - dx0;
        float sd = (d != 0.0f) ? d : 1.0f;
        float interior = dy0 + (dy1 - dy0) * (x - dx0) / sd;
        float r = (x <= cdfT[0]) ? 0.0f : ((x >= cdfT[255]) ? 1.0f : interior);
        pmout[i] = r;
    }
}

// ---------------------------------------------------------------------------
// Kernel 3: per-pixel HSV -> V remap via pxmap -> RGB, streaming b128.
// ---------------------------------------------------------------------------
__device__ __forceinline__ float3 hm_pixel(float r0, float g0, float b0, const float* pm) {
    float r = (r0 + 1.0f) * 127.0f / 255.0f;
    float g = (g0 + 1.0f) * 127.0f / 255.0f;
    float b = (b0 + 1.0f) * 127.0f / 255.0f;
    float v  = fmaxf(fmaxf(r, g), b);
    float mn = fminf(fminf(r, g), b);
    float rng = v - mn;
    float s = (v > 0.0f) ? (rng / v) : 0.0f;
    float safe = (rng > 0.0f) ? rng : 1.0f;
    float norm = 1.0f / (6.0f * safe);
    float h = (v == r) ? (norm * (g - b))
            : ((v == g) ? (norm * (b - r) + 2.0f / 6.0f)
                        : (norm * (r - g) + 4.0f / 6.0f));
    h = (rng == 0.0f) ? 0.0f : h;
    h += (h < 0.0f) ? 1.0f : 0.0f;

    // interpolate(dx = j/255, dy = pm, x = v)
    float x = v, nv;
    if (x <= 0.0f)      nv = pm[0];
    else if (x >= 1.0f) nv = pm[255];
    else {
        int k = (int)(x * 255.0f);
        int i1 = (k > 0) ? (k - 1) : 0;     // first j with j/255 > x (<=3 steps)
        while (i1 < 256 && ((float)i1 / 255.0f) <= x) ++i1;
        int ind1 = iminc(i1, 255);
        int ind0 = imaxc(ind1 - 1, 0);
        float dx0 = (float)ind0 / 255.0f, dx1 = (float)ind1 / 255.0f;
        float dy0 = pm[ind0], dy1 = pm[ind1];
        float d = dx1 - dx0;
        float sd = (d != 0.0f) ? d : 1.0f;
        nv = dy0 + (dy1 - dy0) * (x - dx0) / sd;
    }

    float dh = h * 6.0f;
    float fi = floorf(dh);
    float f  = dh - fi;
    float p  = nv * (1.0f - s);
    float q  = nv * (1.0f - f * s);
    float t  = nv * (1.0f - (1.0f - f) * s);
    int ii = ((int)fi) % 6;
    float R, G, B;
    if      (ii == 0) { R = nv; G = t;  B = p;  }
    else if (ii == 1) { R = q;  G = nv; B = p;  }
    else if (ii == 2) { R = p;  G = nv; B = t;  }
    else if (ii == 3) { R = p;  G = q;  B = nv; }
    else if (ii == 4) { R = t;  G = p;  B = nv; }
    else              { R = nv; G = p;  B = q;  }
    float3 o;
    o.x = R * 255.0f / 127.0f - 1.0f;
    o.y = G * 255.0f / 127.0f - 1.0f;
    o.z = B * 255.0f / 127.0f - 1.0f;
    return o;
}

__global__ void hm_apply(const float* __restrict__ src, const float* __restrict__ pmg,
                         float* __restrict__ out, int ngroups) {
    __shared__ float pm[256];
    for (int i = threadIdx.x; i < 256; i += blockDim.x) pm[i] = pmg[i];
    __syncthreads();

    int stride = gridDim.x * blockDim.x;
    for (int g = blockIdx.x * blockDim.x + threadIdx.x; g < ngroups; g += stride) {
        size_t base = (size_t)g * 12;
        const float4* ps = (const float4*)(src + base);
        __builtin_prefetch(src + base + (size_t)stride * 12, 0, 1);  // global_prefetch_b8
        float4 a = ps[0], b = ps[1], c = ps[2];
        float3 o0 = hm_pixel(a.x, a.y, a.z, pm);
        float3 o1 = hm_pixel(a.w, b.x, b.y, pm);
        float3 o2 = hm_pixel(b.z, b.w, c.x, pm);
        float3 o3 = hm_pixel(c.y, c.z, c.w, pm);
        float4* po = (float4*)(out + base);
        po[0] = make_float4(o0.x, o0.y, o0.z, o1.x);
        po[1] = make_float4(o1.y, o1.z, o2.x, o2.y);
        po[2] = make_float4(o2.z, o3.x, o3.y, o3.z);
    }
}

// ---------------------------------------------------------------------------
extern "C" void kernel_launch(void* const* d_in, const int* in_sizes, int n_in,
                              void* d_out, int out_size, void* d_ws, size_t ws_size,
                              hipStream_t stream) {
    const float* src = (const float*)d_in[0];
    const float* tgt = (const float*)d_in[1];
    float* out = (float*)d_out;

    unsigned* hist = (unsigned*)d_ws;            // 512 u32
    float* pm = (float*)d_ws + 2 * NBINS;        // 256 f32

    int npix = in_sizes[0] / 3;                  // 2160*3840
    int ngroups = npix / 4;                      // 4 pixels (48B) per work item

    hm_zero<<<2, 256, 0, stream>>>(hist);
    hm_hist<<<2048, 256, 0, stream>>>(src, tgt, hist, ngroups);
    hm_cdf_map<<<1, 32, 0, stream>>>(hist, pm, npix);
    hm_apply<<<4096, 256, 0, stream>>>(src, pm, out, ngroups);
}